// LNN_29154238005966
// MI455X (gfx1250) — compile-verified
//
#include <hip/hip_runtime.h>
#include <hip/hip_bf16.h>

// LNN forward: MLP grad + 3 Hessian rows (HVPs) + per-sample 3x3 solve.
// Dominant work: 8 GEMMs of B x 256 x 256 (fp32) -> V_WMMA_F32_16X16X4_F32.
// M-tile = 64 rows/WG (4 sub-tiles per wave sharing one B fragment), W1 is
// pre-transposed once so all B reads are coalesced from L2.

typedef __attribute__((ext_vector_type(2))) float v2f;
typedef __attribute__((ext_vector_type(8))) float v8f;

#define HID 256
#define LDA 260     // padded LDS row stride (floats) to avoid bank conflicts
#define MT  64      // rows per workgroup tile

// ---------------------------------------------------------------------------
// h0 = tanh(W0 x + b0) : K=7, trivial FLOPs -> VALU
__global__ void k_h0(const float* __restrict__ x, const float* __restrict__ W0,
                     const float* __restrict__ b0, float* __restrict__ H0,
                     int base, int nB)
{
    int tid = blockIdx.x * 256 + threadIdx.x;
    int b = tid >> 8;
    int h = tid & 255;
    if (b >= nB) return;
    const float* xr = x + (size_t)(base + b) * 7;
    const float* wr = W0 + (size_t)h * 7;
    float a = b0[h];
#pragma unroll
    for (int d = 0; d < 7; ++d) a += wr[d] * xr[d];
    H0[(size_t)b * HID + h] = tanhf(a);
}

// ---------------------------------------------------------------------------
// W1T = W1^T (once; 256 KB, makes all GEMM B-reads coalesced)
__global__ void k_transpose(const float* __restrict__ W1, float* __restrict__ W1T)
{
    int i = blockIdx.x * 256 + threadIdx.x;   // 65536 elements
    int r = i >> 8;
    int c = i & 255;
    W1T[(size_t)c * HID + r] = W1[i];
}

// ---------------------------------------------------------------------------
// Fused WMMA GEMM over K=256 with per-mode prologue/epilogue.
// WG = 512 threads = 16 waves. Tile: 64 rows x 256 cols.
// Wave w -> cols [16w,16w+16), 4 row sub-tiles of 16 sharing each B fragment.
// Bmat is passed pre-oriented so B[k,n] = Bmat[k*256+n] for every mode.
// MODE 0: OUT0 = H1 = tanh(H0 @ W1^T + b1); OUT1 = G1 = (1-H1^2)*w2
// MODE 1: OUT0 = T  = G1 @ W1
// MODE 2: A = (1-H0^2)*W0[:,j];  OUT0 = DG1 = -2*H1*(1-H1^2)*(A@W1^T)*w2
// MODE 3: OUT0 = DS0 = -2*H0*((1-H0^2)*W0[:,j])*T + (1-H0^2)*(DG1@W1)
template <int MODE>
__global__ void __launch_bounds__(512)
gemm256(const float* __restrict__ Asrc, const float* __restrict__ Bmat,
        const float* __restrict__ b1, const float* __restrict__ w2,
        const float* __restrict__ W0, int jcol,
        const float* __restrict__ H0e, const float* __restrict__ H1e,
        const float* __restrict__ Te,
        float* __restrict__ O0, float* __restrict__ O1, int nB)
{
    __shared__ float As[MT * LDA];
    const int row0 = blockIdx.x * MT;
    const int lane = threadIdx.x & 31;
    const int wv   = threadIdx.x >> 5;
    const int lm   = lane & 15;        // M (or N) index within 16
    const int lh   = lane >> 4;        // K half selector

    // Stage A tile (with prologue transform) into LDS
    for (int i = threadIdx.x; i < MT * HID; i += 512) {
        int r = i >> 8;
        int k = i & 255;
        int gr = row0 + r;
        float a = 0.f;
        if (gr < nB) a = Asrc[(size_t)gr * HID + k];
        if (MODE == 2) { float h0 = a; a = (1.f - h0 * h0) * W0[k * 7 + jcol]; }
        As[r * LDA + k] = a;
    }
    __syncthreads();

    const int n = wv * 16 + lm;                 // output column (hidden idx)

    v8f c0 = {}, c1 = {}, c2 = {}, c3 = {};
#pragma unroll 4
    for (int kb = 0; kb < HID; kb += 4) {
        int kk = kb + 2 * lh;
        v2f b;
        b.x = Bmat[(size_t)kk * HID + n];
        b.y = Bmat[(size_t)(kk + 1) * HID + n];
        v2f a0 = *reinterpret_cast<const v2f*>(&As[(0 * 16 + lm) * LDA + kk]);
        v2f a1 = *reinterpret_cast<const v2f*>(&As[(1 * 16 + lm) * LDA + kk]);
        v2f a2 = *reinterpret_cast<const v2f*>(&As[(2 * 16 + lm) * LDA + kk]);
        v2f a3 = *reinterpret_cast<const v2f*>(&As[(3 * 16 + lm) * LDA + kk]);
        c0 = __builtin_amdgcn_wmma_f32_16x16x4_f32(false, a0, false, b, (short)0, c0, false, false);
        c1 = __builtin_amdgcn_wmma_f32_16x16x4_f32(false, a1, false, b, (short)0, c1, false, false);
        c2 = __builtin_amdgcn_wmma_f32_16x16x4_f32(false, a2, false, b, (short)0, c2, false, false);
        c3 = __builtin_amdgcn_wmma_f32_16x16x4_f32(false, a3, false, b, (short)0, c3, false, false);
    }

    // Epilogue: VGPR v, lanes 0-15 -> M=v ; lanes 16-31 -> M=8+v ; N = lane%16
    v8f cs[4] = {c0, c1, c2, c3};
#pragma unroll
    for (int s = 0; s < 4; ++s) {
        const int mbase = row0 + s * 16 + 8 * lh;
#pragma unroll
        for (int v = 0; v < 8; ++v) {
            int m = mbase + v;
            if (m >= nB) continue;
            size_t idx = (size_t)m * HID + n;
            float acc = cs[s][v];
            if (MODE == 0) {
                float t = tanhf(acc + b1[n]);
                O0[idx] = t;
                O1[idx] = (1.f - t * t) * w2[n];
            } else if (MODE == 1) {
                O0[idx] = acc;
            } else if (MODE == 2) {
                float h1 = H1e[idx];
                O0[idx] = -2.f * h1 * (1.f - h1 * h1) * acc * w2[n];
            } else { // MODE 3
                float h0 = H0e[idx];
                float u = 1.f - h0 * h0;
                O0[idx] = -2.f * h0 * (u * W0[n * 7 + jcol]) * Te[idx] + u * acc;
            }
        }
    }
}

// ---------------------------------------------------------------------------
// out7[b,:] = (useH0 ? (1-H0^2).*S : S)[b,:] @ W0   (256 -> 7 reduction, VALU)
__global__ void k_rowred(const float* __restrict__ S, const float* __restrict__ H0,
                         const float* __restrict__ W0, float* __restrict__ out,
                         int outStride, int useH0, int nB)
{
    int b = blockIdx.x * blockDim.x + threadIdx.x;
    if (b >= nB) return;
    float acc[7] = {0.f, 0.f, 0.f, 0.f, 0.f, 0.f, 0.f};
    const float* sr = S + (size_t)b * HID;
    const float* hr = H0 + (size_t)b * HID;
    for (int k = 0; k < HID; ++k) {
        float s = sr[k];
        if (useH0) { float h0 = hr[k]; s *= (1.f - h0 * h0); }
        const float* wr = W0 + k * 7;
#pragma unroll
        for (int d = 0; d < 7; ++d) acc[d] += s * wr[d];
    }
    float* o = out + (size_t)b * outStride;
#pragma unroll
    for (int d = 0; d < 7; ++d) o[d] = acc[d];
}

// ---------------------------------------------------------------------------
// Per-sample assembly + 3x3 Cramer solve
__global__ void k_final(const float* __restrict__ x, const float* __restrict__ gradL,
                        const float* __restrict__ Hrows, float* __restrict__ out,
                        int base, int nB)
{
    int b = blockIdx.x * blockDim.x + threadIdx.x;
    if (b >= nB) return;
    const float* xp = x + (size_t)(base + b) * 7;
    float xv[7];
#pragma unroll
    for (int d = 0; d < 7; ++d) xv[d] = xp[d];
    const float* g  = gradL + (size_t)b * 7;
    const float* Hr = Hrows + (size_t)b * 21;
    float x1 = xv[2], x2 = xv[3];
    float dq0 = xv[4], dq1 = xv[5], dq2 = xv[6];

    float Hq[3][3], rhs[3];
#pragma unroll
    for (int i = 0; i < 3; ++i) {
        float jr0 = Hr[i * 7 + 0], jr1 = Hr[i * 7 + 1];
        float jx1 = Hr[i * 7 + 2], jx2 = Hr[i * 7 + 3];
        float jth = -x2 * jx1 + x1 * jx2;
#pragma unroll
        for (int j = 0; j < 3; ++j) Hq[i][j] = Hr[i * 7 + 4 + j];
        Hq[i][i] += 1e-6f;
        float dL = (i == 0) ? g[0] : (i == 1) ? g[1] : (-x2 * g[2] + x1 * g[3]);
        rhs[i] = dL - (jr0 * dq0 + jr1 * dq1 + jth * dq2);
    }
    float a00 = Hq[0][0], a01 = Hq[0][1], a02 = Hq[0][2];
    float a10 = Hq[1][0], a11 = Hq[1][1], a12 = Hq[1][2];
    float a20 = Hq[2][0], a21 = Hq[2][1], a22 = Hq[2][2];
    float r0 = rhs[0], r1 = rhs[1], r2 = rhs[2];
    float det = a00 * (a11 * a22 - a12 * a21)
              - a01 * (a10 * a22 - a12 * a20)
              + a02 * (a10 * a21 - a11 * a20);
    float inv = 1.f / det;
    float d0 = r0 * (a11 * a22 - a12 * a21) - a01 * (r1 * a22 - a12 * r2) + a02 * (r1 * a21 - a11 * r2);
    float d1 = a00 * (r1 * a22 - a12 * r2) - r0 * (a10 * a22 - a12 * a20) + a02 * (a10 * r2 - r1 * a20);
    float d2 = a00 * (a11 * r2 - r1 * a21) - a01 * (a10 * r2 - r1 * a20) + r0 * (a10 * a21 - a11 * a20);

    float* o = out + (size_t)(base + b) * 7;
    o[0] = dq0;
    o[1] = dq1;
    o[2] = -x2 * dq2;
    o[3] = x1 * dq2;
    o[4] = d0 * inv;
    o[5] = d1 * inv;
    o[6] = d2 * inv;
}

// ---------------------------------------------------------------------------
extern "C" void kernel_launch(void* const* d_in, const int* in_sizes, int n_in,
                              void* d_out, int out_size, void* d_ws, size_t ws_size,
                              hipStream_t stream)
{
    const float* x  = (const float*)d_in[0];
    const float* W0 = (const float*)d_in[1];
    const float* b0 = (const float*)d_in[2];
    const float* W1 = (const float*)d_in[3];
    const float* b1 = (const float*)d_in[4];
    const float* w2 = (const float*)d_in[5];
    float* out = (float*)d_out;
    int Btot = in_sizes[0] / 7;

    // ws: [W1T: 64K floats][H0|H1|T|G: CH*256 each][gradL: CH*7][Hrows: CH*21]
    const size_t w1tBytes = (size_t)HID * HID * sizeof(float);
    const size_t perRowBytes = (size_t)(4 * HID + 28) * sizeof(float);
    size_t avail = (ws_size > w1tBytes) ? (ws_size - w1tBytes) : 0;
    size_t fit = avail / perRowBytes;
    int CH = (fit > 32768) ? 32768 : (int)(fit & ~(size_t)63);
    if (CH < 64) CH = 64;
    if (CH > Btot) CH = (Btot + 63) & ~63;

    float* W1T   = (float*)d_ws;
    float* H0b   = W1T + (size_t)HID * HID;
    float* H1b   = H0b + (size_t)CH * HID;
    float* Tb    = H1b + (size_t)CH * HID;
    float* Gb    = Tb  + (size_t)CH * HID;
    float* gradL = Gb  + (size_t)CH * HID;
    float* Hrows = gradL + (size_t)CH * 7;

    k_transpose<<<HID * HID / 256, 256, 0, stream>>>(W1, W1T);

    for (int base = 0; base < Btot; base += CH) {
        int nB = (Btot - base < CH) ? (Btot - base) : CH;
        int gw = (nB + MT - 1) / MT;
        int gb = (nB + 255) / 256;

        k_h0<<<(nB * 256 + 255) / 256, 256, 0, stream>>>(x, W0, b0, H0b, base, nB);

        // H1 = tanh(H0 @ W1^T + b1), G1 = (1-H1^2)*w2   (B = W1T, coalesced)
        gemm256<0><<<gw, 512, 0, stream>>>(H0b, W1T, b1, w2, W0, 0,
                                           nullptr, nullptr, nullptr, H1b, Gb, nB);
        // T = G1 @ W1
        gemm256<1><<<gw, 512, 0, stream>>>(Gb, W1, b1, w2, W0, 0,
                                           nullptr, nullptr, nullptr, Tb, nullptr, nB);
        // gradL = ((1-H0^2).*T) @ W0
        k_rowred<<<gb, 256, 0, stream>>>(Tb, H0b, W0, gradL, 7, 1, nB);

        for (int tj = 0; tj < 3; ++tj) {
            int jcol = 4 + tj;
            // DG1 = -2*H1*(1-H1^2)*((dh0)@W1^T)*w2   (B = W1T)
            gemm256<2><<<gw, 512, 0, stream>>>(H0b, W1T, b1, w2, W0, jcol,
                                               nullptr, H1b, nullptr, Gb, nullptr, nB);
            // DS0 = -2*H0*dh0*T + (1-H0^2)*(DG1@W1)  (in-place into Gb: A staged in LDS)
            gemm256<3><<<gw, 512, 0, stream>>>(Gb, W1, b1, w2, W0, jcol,
                                               H0b, nullptr, Tb, Gb, nullptr, nB);
            // Hrow_tj = DS0 @ W0
            k_rowred<<<gb, 256, 0, stream>>>(Gb, H0b, W0, Hrows + tj * 7, 21, 0, nB);
        }

        k_final<<<gb, 256, 0, stream>>>(x, gradL, Hrows, out, base, nB);
    }
}